// ChargesReadoutBlock_13872744366316
// MI455X (gfx1250) — compile-verified
//
#include <hip/hip_runtime.h>

typedef float v2f __attribute__((ext_vector_type(2)));
typedef float v4f __attribute__((ext_vector_type(4)));
typedef float v8f __attribute__((ext_vector_type(8)));

#define MULC 256
#define FDIM 2304        // 9*MUL features per node
#define NNODES 65536
#define KSTEPS (FDIM/4)  // 576 WMMA k-steps of 4
#define TILES_PER_BLOCK 8
#define THREADS 256

// Map feature column j -> charge component index 0..8
__device__ __forceinline__ int comp_of(int j) {
  if (j < MULC) return 0;
  if (j < 4 * MULC) return 1 + (j - MULC) % 3;
  return 4 + (j - 4 * MULC) % 5;
}

// basew[j] = scale(group) * sum_v W_lin[u(j),v] * W_tp[v]
__global__ void prep_basew(const float* __restrict__ Wl0, const float* __restrict__ Wl1,
                           const float* __restrict__ Wl2, const float* __restrict__ Wt0,
                           const float* __restrict__ Wt1, const float* __restrict__ Wt2,
                           float* __restrict__ basew) {
  int j = blockIdx.x * blockDim.x + threadIdx.x;
  if (j >= FDIM) return;
  const float alpha = rsqrtf(3.0f * 256.0f);   // 1/sqrt(3*MUL)
  const float lin_norm = 1.0f / 16.0f;         // 1/sqrt(MUL)
  const float* Wl;
  const float* Wt;
  int u;
  float s = alpha * lin_norm;
  if (j < MULC)            { Wl = Wl0; Wt = Wt0; u = j; }
  else if (j < 4 * MULC)   { Wl = Wl1; Wt = Wt1; u = (j - MULC) / 3;     s *= rsqrtf(3.0f); }
  else                     { Wl = Wl2; Wt = Wt2; u = (j - 4 * MULC) / 5; s *= rsqrtf(5.0f); }
  float acc = 0.0f;
  for (int v = 0; v < 256; ++v) acc += Wl[u * 256 + v] * Wt[v];
  basew[j] = s * acc;
}

// One wave owns a 16-node tile: S(16x16) = nf_tile(16x2304) @ Wmat(2304x16)
// via V_WMMA_F32_16X16X4_F32, then energy[z] = sum_{k<9} S[z,k]*charges[z,k].
__global__ __launch_bounds__(THREADS) void energy_main(
    const float* __restrict__ nf, const float* __restrict__ charges,
    const float* __restrict__ basew, float* __restrict__ out) {
  // LDS table: per feature column j store (basew[j], comp(j) bits). 18 KB.
  __shared__ float wtab[2 * FDIM];
  for (int j = threadIdx.x; j < FDIM; j += THREADS) {
    wtab[2 * j]     = basew[j];
    wtab[2 * j + 1] = __int_as_float(comp_of(j));
  }
  __syncthreads();

  const int lane = threadIdx.x & 31;
  const int wave = threadIdx.x >> 5;
  const int khi  = lane >> 4;   // selects K pair {0,1} vs {2,3}
  const int n    = lane & 15;   // N column (and A-row index for loads)
  const int z0   = (blockIdx.x * TILES_PER_BLOCK + wave) * 16;

  // A stream: lane reads row (z0+n), bytes [ (4kk+2*khi)*4 , +8 ) each step.
  const v2f* __restrict__ ap = (const v2f*)(nf + (size_t)(z0 + n) * FDIM) + khi;
  // B source: v4f at index 2*kk+khi -> (basew[j], comp[j], basew[j+1], comp[j+1])
  const v4f* __restrict__ wp = ((const v4f*)wtab) + khi;

  v8f acc = {};
  for (int kko = 0; kko < KSTEPS; kko += 8) {
    __builtin_prefetch((const void*)(ap + 40), 0, 0);  // ~320B ahead of stream
#pragma unroll
    for (int ku = 0; ku < 8; ++ku) {
      v2f a = __builtin_nontemporal_load(ap);          // streaming, TH=NT
      v4f q = *wp;                                     // LDS broadcast (2 addrs/wave)
      v2f b;
      b.x = (__float_as_int(q.y) == n) ? q.x : 0.0f;   // Wmat[j,   n]
      b.y = (__float_as_int(q.w) == n) ? q.z : 0.0f;   // Wmat[j+1, n]
      acc = __builtin_amdgcn_wmma_f32_16x16x4_f32(false, a, false, b,
                                                  (short)0, acc, false, false);
      ap += 2;
      wp += 2;
    }
  }

  // C/D layout: acc[r] = S[r + 8*khi, n] for this lane. Contract with charges.
#pragma unroll
  for (int r = 0; r < 8; ++r) {
    int z = z0 + r + 8 * khi;
    float t = (n < 9) ? acc[r] * charges[(size_t)z * 9 + n] : 0.0f;
    t += __shfl_xor(t, 1, 16);
    t += __shfl_xor(t, 2, 16);
    t += __shfl_xor(t, 4, 16);
    t += __shfl_xor(t, 8, 16);
    if (n == 0) out[z] = t;
  }
}

extern "C" void kernel_launch(void* const* d_in, const int* in_sizes, int n_in,
                              void* d_out, int out_size, void* d_ws, size_t ws_size,
                              hipStream_t stream) {
  (void)in_sizes; (void)n_in; (void)out_size; (void)ws_size;
  const float* nf  = (const float*)d_in[0];
  const float* ch  = (const float*)d_in[1];
  const float* Wl0 = (const float*)d_in[2];
  const float* Wl1 = (const float*)d_in[3];
  const float* Wl2 = (const float*)d_in[4];
  const float* Wt0 = (const float*)d_in[5];
  const float* Wt1 = (const float*)d_in[6];
  const float* Wt2 = (const float*)d_in[7];
  float* out   = (float*)d_out;
  float* basew = (float*)d_ws;  // 2304 floats of scratch

  prep_basew<<<(FDIM + 255) / 256, 256, 0, stream>>>(Wl0, Wl1, Wl2, Wt0, Wt1, Wt2, basew);
  energy_main<<<NNODES / (16 * TILES_PER_BLOCK), THREADS, 0, stream>>>(nf, ch, basew, out);
}